// SympatheticCircuit_61667140436067
// MI455X (gfx1250) — compile-verified
//
#include <hip/hip_runtime.h>
#include <hip/hip_bf16.h>
#include <math.h>

typedef float v8f __attribute__((ext_vector_type(8)));
typedef float v2f __attribute__((ext_vector_type(2)));

#define BB 4
#define TT 64
#define FF 64
#define UU 64
#define SS 64
#define MM 64

// ---------------------------------------------------------------------------
// Kernel 1: per batch  h = tanh(in @ Wk + bk);  out = h @ Wv + bv
// One 512-thread block per batch (16 waves = 16 of the 16x16 output tiles).
// Uses v_wmma_f32_16x16x4_f32, K unrolled (16 WMMA per tile per GEMM).
// ---------------------------------------------------------------------------
__global__ __launch_bounds__(512)
void es_gemm_kernel(const float* __restrict__ inputs,
                    const float* __restrict__ Wk, const float* __restrict__ bk,
                    const float* __restrict__ Wv, const float* __restrict__ bv,
                    float* __restrict__ h_g, float* __restrict__ out_g)
{
    __shared__ float sh_h[TT * UU];          // 16 KB

    const int b    = blockIdx.x;
    const int tid  = threadIdx.x;
    const int wave = tid >> 5;
    const int lane = tid & 31;
    const int row  = lane & 15;
    const int half = lane >> 4;
    const int m0   = (wave >> 2) << 4;
    const int n0   = (wave & 3) << 4;

    const float* A = inputs + b * TT * FF;

    // --- stage 1: C = inputs[b] @ Wk ---
    v8f acc = {0.f,0.f,0.f,0.f,0.f,0.f,0.f,0.f};
#pragma unroll
    for (int k = 0; k < FF; k += 4) {
        const int ka = k + 2 * half;
        v2f a, bm;
        a.x  = A[(m0 + row) * FF + ka];
        a.y  = A[(m0 + row) * FF + ka + 1];
        bm.x = Wk[ka * UU + n0 + row];
        bm.y = Wk[(ka + 1) * UU + n0 + row];
        acc = __builtin_amdgcn_wmma_f32_16x16x4_f32(false, a, false, bm,
                                                    (short)0, acc, false, false);
    }
    const float bkv = bk[n0 + row];
#pragma unroll
    for (int j = 0; j < 8; ++j) {
        const int m = m0 + j + 8 * half;
        const float hv = tanhf(acc[j] + bkv);
        sh_h[m * UU + n0 + row] = hv;
        h_g[b * TT * UU + m * UU + n0 + row] = hv;
    }
    __syncthreads();

    // --- stage 2: out = h @ Wv + bv ---
    v8f acc2 = {0.f,0.f,0.f,0.f,0.f,0.f,0.f,0.f};
#pragma unroll
    for (int k = 0; k < UU; k += 4) {
        const int ka = k + 2 * half;
        v2f a, bm;
        a.x  = sh_h[(m0 + row) * UU + ka];
        a.y  = sh_h[(m0 + row) * UU + ka + 1];
        bm.x = Wv[ka * FF + n0 + row];
        bm.y = Wv[(ka + 1) * FF + n0 + row];
        acc2 = __builtin_amdgcn_wmma_f32_16x16x4_f32(false, a, false, bm,
                                                     (short)0, acc2, false, false);
    }
    const float bvv = bv[n0 + row];
#pragma unroll
    for (int j = 0; j < 8; ++j) {
        const int m = m0 + j + 8 * half;
        out_g[b * TT * FF + m * FF + n0 + row] = acc2[j] + bvv;
    }
}

// ---------------------------------------------------------------------------
// Kernel 2: per batch — score / argmax path / flat / attention / distance.
// Exploits tanh monotonicity: the double-gather double-max collapses to
// extreme products of h[t,*] (selected by idx_t) and w_s[*] (selected by idx_f).
// ---------------------------------------------------------------------------
__global__ __launch_bounds__(256)
void path_mem_kernel(const float* __restrict__ inputs,
                     const float* __restrict__ w_s,
                     const float* __restrict__ mem_keys,
                     const float* __restrict__ mem_vals,
                     const float* __restrict__ h_g,
                     const float* __restrict__ out_g,
                     float* __restrict__ dist_g,
                     float* __restrict__ imp_g)
{
    __shared__ float sh_h[TT * UU];     // 16 KB
    __shared__ float sh_o[TT * FF];     // 16 KB
    __shared__ float sh_s[TT * FF];     // 16 KB  (score, then flat)
    __shared__ int   sh_idxf[FF];
    __shared__ int   sh_idxt[TT];
    __shared__ float sh_maxA[TT], sh_minA[TT];
    __shared__ float sh_red[256];
    __shared__ float sh_e[MM];
    __shared__ float sh_tgt[FF];
    __shared__ float sh_ws[SS];
    __shared__ float sh_scal[4];        // maxW, minW, logit-max, exp-sum

    const int b = blockIdx.x;
    const int tid = threadIdx.x;

    for (int i = tid; i < TT * UU; i += 256) sh_h[i] = h_g[b * TT * UU + i];
    for (int i = tid; i < TT * FF; i += 256) sh_o[i] = out_g[b * TT * FF + i];
    if (tid < SS) sh_ws[tid] = w_s[tid];
    __syncthreads();

    const float wlast = sh_ws[SS - 1];

    // score[t,f] = sum_u tanh(h[t,u] * out[t,f] * w_s[-1])
    for (int i = tid; i < TT * FF; i += 256) {
        const int t = i >> 6;
        const float c = sh_o[i] * wlast;
        const float* hr = &sh_h[t * UU];
        float s = 0.f;
#pragma unroll 4
        for (int u = 0; u < UU; ++u) s += tanhf(hr[u] * c);
        sh_s[i] = s;
    }
    __syncthreads();

    // idx_f[f] = argmax_t score[t,f];  idx_t[t] = argmax_f score[t,f]  (first max)
    if (tid < FF) {
        const int f = tid;
        float best = sh_s[f]; int bi = 0;
        for (int t = 1; t < TT; ++t) {
            const float v = sh_s[t * FF + f];
            if (v > best) { best = v; bi = t; }
        }
        sh_idxf[f] = bi;
    } else if (tid < 128) {
        const int t = tid - 64;
        float best = sh_s[t * FF]; int bi = 0;
        for (int f = 1; f < FF; ++f) {
            const float v = sh_s[t * FF + f];
            if (v > best) { best = v; bi = f; }
        }
        sh_idxt[t] = bi;
    }
    __syncthreads();

    if (tid == 0) {
        float mw = -3.402823466e38f, nw = 3.402823466e38f;
        for (int j = 0; j < FF; ++j) {
            const float v = sh_ws[sh_idxf[j]];
            mw = fmaxf(mw, v); nw = fminf(nw, v);
        }
        sh_scal[0] = mw; sh_scal[1] = nw;
    }
    if (tid < TT) {
        const int t = tid;
        float mx = -3.402823466e38f, mn = 3.402823466e38f;
        for (int i = 0; i < TT; ++i) {
            const float v = sh_h[t * UU + sh_idxt[i]];
            mx = fmaxf(mx, v); mn = fminf(mn, v);
        }
        sh_maxA[t] = mx; sh_minA[t] = mn;
    }
    __syncthreads();

    const float maxW = sh_scal[0], minW = sh_scal[1];
    // flat[t,f] = tanh(max over selected (u,s) of h*out*w)
    for (int i = tid; i < TT * FF; i += 256) {
        const int t = i >> 6;
        const float p  = sh_o[i];
        const float M1 = fmaxf(sh_maxA[t] * maxW, sh_minA[t] * minW);
        const float M2 = fminf(sh_maxA[t] * minW, sh_minA[t] * maxW);
        sh_s[i] = tanhf((p >= 0.f) ? p * M1 : p * M2);
    }
    __syncthreads();

    // logits[m] = flat . mem_keys[m]  (4 threads per m)
    {
        const int m = tid >> 2, part = tid & 3;
        const float* kr = mem_keys + m * (TT * FF) + part * 1024;
        const float* fr = sh_s + part * 1024;
        float s = 0.f;
        for (int i = 0; i < 1024; ++i) s += fr[i] * kr[i];
        sh_red[tid] = s;
    }
    __syncthreads();
    if (tid < MM)
        sh_e[tid] = sh_red[4 * tid] + sh_red[4 * tid + 1] +
                    sh_red[4 * tid + 2] + sh_red[4 * tid + 3];
    __syncthreads();
    if (tid == 0) {
        float mx = sh_e[0];
        for (int m = 1; m < MM; ++m) mx = fmaxf(mx, sh_e[m]);
        sh_scal[2] = mx;
    }
    __syncthreads();
    if (tid < MM) sh_e[tid] = expf(sh_e[tid] - sh_scal[2]);
    __syncthreads();
    if (tid == 0) {
        float s = 0.f;
        for (int m = 0; m < MM; ++m) s += sh_e[m];
        sh_scal[3] = s;
    }
    __syncthreads();
    const float inv = 1.0f / sh_scal[3];   // importance = max(softmax) = 1/sum

    if (tid < FF) {
        float s = 0.f;
        for (int m = 0; m < MM; ++m) s += sh_e[m] * mem_vals[m * FF + tid];
        sh_tgt[tid] = s * inv;
    }
    __syncthreads();

    if (tid < TT) {
        const int t = tid;
        const float* xr = inputs + b * TT * FF + t * FF;
        float ss = 0.f;
        for (int f = 0; f < FF; ++f) {
            const float d = xr[f] - sh_tgt[f];
            ss += d * d;
        }
        const float nrm = sqrtf(ss);
        const float hs  = fminf(fmaxf(0.2f * nrm + 0.5f, 0.f), 1.f);
        dist_g[b * TT + t] = 0.5f - hs;
        imp_g [b * TT + t] = inv;
    }
}

// ---------------------------------------------------------------------------
extern "C" void kernel_launch(void* const* d_in, const int* in_sizes, int n_in,
                              void* d_out, int out_size, void* d_ws, size_t ws_size,
                              hipStream_t stream)
{
    const float* inputs   = (const float*)d_in[0];
    const float* Wk       = (const float*)d_in[1];
    const float* bk       = (const float*)d_in[2];
    const float* Wv       = (const float*)d_in[3];
    const float* bv       = (const float*)d_in[4];
    const float* w_s      = (const float*)d_in[5];
    const float* mem_keys = (const float*)d_in[6];
    const float* mem_vals = (const float*)d_in[7];

    float* out    = (float*)d_out;
    float* dist_g = out;                    // [B,T,1]  -> 256
    float* imp_g  = out + BB * TT;          // [B,T,1]  -> 256
    float* out_g  = out + 2 * BB * TT;      // [B,T,F]  -> 16384

    float* h_g = (float*)d_ws;              // [B,T,U] scratch (64 KB)

    es_gemm_kernel<<<BB, 512, 0, stream>>>(inputs, Wk, bk, Wv, bv, h_g, out_g);
    path_mem_kernel<<<BB, 256, 0, stream>>>(inputs, w_s, mem_keys, mem_vals,
                                            h_g, out_g, dist_g, imp_g);
}